// AttentionModule_41403484733949
// MI455X (gfx1250) — compile-verified
//
#include <hip/hip_runtime.h>
#include <hip/hip_bf16.h>

typedef __attribute__((ext_vector_type(2))) float v2f;
typedef __attribute__((ext_vector_type(8))) float v8f;

#define DFEAT 128   // feature dim (known from reference)

// ---------------------------------------------------------------------------
// Zero-init output + accumulators (harness poisons buffers with 0xAA).
// ---------------------------------------------------------------------------
__global__ void zero3_kernel(float* a, int na, float* b, int nb, float* c, int nc) {
    int stride = gridDim.x * blockDim.x;
    int tid = blockIdx.x * blockDim.x + threadIdx.x;
    for (int i = tid; i < na; i += stride) a[i] = 0.0f;
    for (int i = tid; i < nb; i += stride) b[i] = 0.0f;
    for (int i = tid; i < nc; i += stride) c[i] = 0.0f;
}

// ---------------------------------------------------------------------------
// Pass 1: per-segment sum of raw features + counts.
// One wave handles `chunk` consecutive nodes; 32 lanes x float4 = 128 dims.
// segment_ids are sorted -> run-length accumulate in registers, atomics only
// at segment boundaries.
// ---------------------------------------------------------------------------
__global__ void seg_feature_sum_kernel(const float4* __restrict__ feats4,
                                       const long long* __restrict__ seg,
                                       float* __restrict__ segacc,
                                       float* __restrict__ segcnt,
                                       int N, int chunk) {
    int wave = (int)((blockIdx.x * blockDim.x + threadIdx.x) >> 5);
    int lane = (int)(threadIdx.x & 31);
    int base = wave * chunk;
    if (base >= N) return;
    int end = base + chunk; if (end > N) end = N;

    float ax = 0.f, ay = 0.f, az = 0.f, aw = 0.f;
    int runlen = 0;
    long long prev = -1;

    for (int n = base; n < end; ++n) {
        long long s = seg[n];                 // uniform across wave (scalar load)
        if (s != prev) {
            if (runlen > 0) {
                float* dst = segacc + prev * DFEAT + lane * 4;
                atomicAdd(dst + 0, ax); atomicAdd(dst + 1, ay);
                atomicAdd(dst + 2, az); atomicAdd(dst + 3, aw);
                if (lane == 0) atomicAdd(segcnt + prev, (float)runlen);
            }
            ax = ay = az = aw = 0.f; runlen = 0; prev = s;
        }
        float4 f = feats4[n * (DFEAT / 4) + lane];   // coalesced 512B row load
        ax += f.x; ay += f.y; az += f.z; aw += f.w;
        ++runlen;
    }
    if (runlen > 0) {
        float* dst = segacc + prev * DFEAT + lane * 4;
        atomicAdd(dst + 0, ax); atomicAdd(dst + 1, ay);
        atomicAdd(dst + 2, az); atomicAdd(dst + 3, aw);
        if (lane == 0) atomicAdd(segcnt + prev, (float)runlen);
    }
}

// ---------------------------------------------------------------------------
// Pass 2: ctx = tanh( (segacc/cnt) @ W ) using V_WMMA_F32_16X16X4_F32.
// One wave computes one 16(M) x 16(N) tile, K=128 in 32 WMMA steps.
// fp32 A 16x4 layout: lanes 0-15 M=0..15 {K=k0,k0+1}; lanes 16-31 {K=k0+2,k0+3}
// fp32 B 4x16 layout: lanes 0-15 N=0..15 {K=k0,k0+1}; lanes 16-31 {K=k0+2,k0+3}
// fp32 C/D 16x16: VGPR r: lanes 0-15 -> (M=r, N=lane); lanes 16-31 -> (M=r+8).
// EXEC is all-ones in every wave that reaches the WMMA (uniform guard only).
// ---------------------------------------------------------------------------
__global__ void ctx_gemm_tanh_kernel(const float* __restrict__ segacc,
                                     const float* __restrict__ segcnt,
                                     const float* __restrict__ Wmat,
                                     float* __restrict__ ctx,
                                     int ntiles_n, int ntiles_total) {
    int wave = (int)((blockIdx.x * blockDim.x + threadIdx.x) >> 5);
    int lane = (int)(threadIdx.x & 31);
    if (wave >= ntiles_total) return;     // uniform per wave

    int mt = wave / ntiles_n;
    int nt = wave - mt * ntiles_n;
    int m0 = mt * 16, n0 = nt * 16;
    int half = lane >> 4;                 // 0: K pair {k0,k0+1}; 1: {k0+2,k0+3}
    int lm   = lane & 15;

    float inv = 1.0f / fmaxf(segcnt[m0 + lm], 1.0f);
    const float* Arow = segacc + (size_t)(m0 + lm) * DFEAT;

    v8f c = {0.f, 0.f, 0.f, 0.f, 0.f, 0.f, 0.f, 0.f};
    #pragma unroll
    for (int k0 = 0; k0 < DFEAT; k0 += 4) {
        int ka = k0 + 2 * half;
        v2f a, b;
        a.x = Arow[ka]     * inv;                 // A = mean matrix row m0+lm
        a.y = Arow[ka + 1] * inv;
        b.x = Wmat[(size_t)ka * DFEAT + n0 + lm]; // B = W[k][n], row-major
        b.y = Wmat[(size_t)(ka + 1) * DFEAT + n0 + lm];
        c = __builtin_amdgcn_wmma_f32_16x16x4_f32(
                /*neg_a=*/false, a, /*neg_b=*/false, b,
                /*c_mod=*/(short)0, c, /*reuse_a=*/false, /*reuse_b=*/false);
    }

    #pragma unroll
    for (int r = 0; r < 8; ++r) {
        int row = r + 8 * half;
        ctx[(size_t)(m0 + row) * DFEAT + n0 + lm] = tanhf(c[r]);
    }
}

// ---------------------------------------------------------------------------
// Pass 3: score_n = <feature_n, ctx[seg_n]>;  rep[seg] += score_n * feature_n.
// Same wave-per-chunk structure as pass 1; ctx rows (5 MB) live in L2.
// ---------------------------------------------------------------------------
__global__ void attn_rep_kernel(const float4* __restrict__ feats4,
                                const long long* __restrict__ seg,
                                const float4* __restrict__ ctx4,
                                float* __restrict__ rep,
                                int N, int chunk) {
    int wave = (int)((blockIdx.x * blockDim.x + threadIdx.x) >> 5);
    int lane = (int)(threadIdx.x & 31);
    int base = wave * chunk;
    if (base >= N) return;
    int end = base + chunk; if (end > N) end = N;

    float ax = 0.f, ay = 0.f, az = 0.f, aw = 0.f;
    long long prev = -1;

    for (int n = base; n < end; ++n) {
        long long s = seg[n];
        if (s != prev) {
            if (prev >= 0) {
                float* dst = rep + prev * DFEAT + lane * 4;
                atomicAdd(dst + 0, ax); atomicAdd(dst + 1, ay);
                atomicAdd(dst + 2, az); atomicAdd(dst + 3, aw);
            }
            ax = ay = az = aw = 0.f; prev = s;
        }
        float4 f = feats4[n * (DFEAT / 4) + lane];
        float4 c = ctx4[(int)s * (DFEAT / 4) + lane];
        float p = f.x * c.x + f.y * c.y + f.z * c.z + f.w * c.w;
        #pragma unroll
        for (int off = 16; off > 0; off >>= 1)    // wave32 butterfly reduce
            p += __shfl_xor(p, off, 32);
        ax += p * f.x; ay += p * f.y; az += p * f.z; aw += p * f.w;
    }
    if (prev >= 0) {
        float* dst = rep + prev * DFEAT + lane * 4;
        atomicAdd(dst + 0, ax); atomicAdd(dst + 1, ay);
        atomicAdd(dst + 2, az); atomicAdd(dst + 3, aw);
    }
}

// ---------------------------------------------------------------------------
extern "C" void kernel_launch(void* const* d_in, const int* in_sizes, int n_in,
                              void* d_out, int out_size, void* d_ws, size_t ws_size,
                              hipStream_t stream) {
    const float*      feats = (const float*)d_in[0];
    const long long*  seg   = (const long long*)d_in[1];
    const float*      Wmat  = (const float*)d_in[3];
    float*            rep   = (float*)d_out;

    int N = in_sizes[1];             // number of nodes (= #segment ids)
    int G = out_size / DFEAT;        // number of segments (10000)

    // Workspace layout: segacc [G*128] | segcnt [G] | ctx [G*128]
    float* segacc = (float*)d_ws;
    float* segcnt = segacc + (size_t)G * DFEAT;
    float* ctx    = segcnt + G;

    int nz = G * DFEAT;
    zero3_kernel<<<1024, 256, 0, stream>>>(rep, nz, segacc, nz, segcnt, G);

    const int chunk = 128;                       // nodes per wave
    int waves  = (N + chunk - 1) / chunk;
    int blocksA = (waves + 7) / 8;               // 8 waves per 256-thread block
    seg_feature_sum_kernel<<<blocksA, 256, 0, stream>>>(
        (const float4*)feats, seg, segacc, segcnt, N, chunk);

    int ntn   = DFEAT / 16;                      // 8 N-tiles
    int tiles = (G / 16) * ntn;                  // 5000 tiles
    int blocksB = (tiles + 7) / 8;
    ctx_gemm_tanh_kernel<<<blocksB, 256, 0, stream>>>(
        segacc, segcnt, Wmat, ctx, ntn, tiles);

    attn_rep_kernel<<<blocksA, 256, 0, stream>>>(
        (const float4*)feats, seg, (const float4*)ctx, rep, N, chunk);
}